// GCNReaonser_45904610459832
// MI455X (gfx1250) — compile-verified
//
#include <hip/hip_runtime.h>
#include <hip/hip_bf16.h>

// GCN reasoner for MI455X (gfx1250, wave32).
// Bottleneck: edge scatter/gather (~6.5 GB random traffic @ 23.3 TB/s).
// GEMMs use V_WMMA_F32_16X16X4_F32 (full f32 precision; FLOPs hidden behind
// the bandwidth-bound aggregation). A-tiles are staged global->LDS with the
// gfx1250 async-to-LDS DMA path, double buffered, synced via ASYNCcnt.

typedef __attribute__((ext_vector_type(2))) float v2f;
typedef __attribute__((ext_vector_type(8))) float v8f;
typedef int v4i __attribute__((vector_size(16)));

#define DIM 128       // D == H == 128
#define LSTRIDE 132   // LDS row stride (floats): 528B rows -> b128-aligned
                      // chunks, conflict-free ds_load_b64 (bank = 4*lane+k)

// ---- gfx1250 async-to-LDS plumbing ----------------------------------------
// Builtin prototype (from clang diagnostic): (v4i AS1*, v4i AS3*, imm, imm).
#define AS1P(p) ((__attribute__((address_space(1))) v4i*)(uintptr_t)(p))
#define AS3P(p) ((__attribute__((address_space(3))) v4i*)(unsigned)(uintptr_t)(p))

#if defined(__has_builtin)
#if __has_builtin(__builtin_amdgcn_global_load_async_to_lds_b128)
#define HAVE_ASYNC_LDS 1
#endif
#if __has_builtin(__builtin_amdgcn_s_wait_asynccnt)
#define HAVE_WAIT_ASYNC 1
#endif
#endif

__device__ __forceinline__ void wait_async0() {
#if defined(HAVE_WAIT_ASYNC)
    __builtin_amdgcn_s_wait_asynccnt(0);
#elif defined(HAVE_ASYNC_LDS)
    asm volatile("s_wait_asynccnt 0x0" ::: "memory");
#endif
}

// ---------------------------------------------------------------- utilities
__device__ __forceinline__ bool keep_mask(unsigned i) {
    // Deterministic PCG-style hash keyed on 42; keep with p = 0.8.
    unsigned s = i * 747796405u + 2891336453u + 42u * 0x9E3779B9u;
    unsigned w = ((s >> ((s >> 28u) + 4u)) ^ s) * 277803737u;
    w = (w >> 22u) ^ w;
    return (w & 0xFFFFFFu) < 13421773u;  // 0.8 * 2^24
}

// ---------------------------------------------------------------- degree / norm
__global__ __launch_bounds__(256) void k_fill1(float* deg, int n) {
    int i = blockIdx.x * blockDim.x + threadIdx.x;
    if (i < n) deg[i] = 1.0f;  // self-loop
}

__global__ __launch_bounds__(256) void k_deg(const int* __restrict__ edst,
                                             float* __restrict__ deg, int e) {
    int i = blockIdx.x * blockDim.x + threadIdx.x;
    if (i < e) unsafeAtomicAdd(&deg[edst[i]], 1.0f);
}

__global__ __launch_bounds__(256) void k_rsqrt(float* d, int n) {
    int i = blockIdx.x * blockDim.x + threadIdx.x;
    if (i < n) d[i] = __frsqrt_rn(d[i]);  // deg >= 1 always
}

// ---------------------------------------------------------------- h0 = query[batch] * x
__global__ __launch_bounds__(256) void k_inith(const float* __restrict__ x,
                                               const float* __restrict__ query,
                                               const int* __restrict__ batch,
                                               float* __restrict__ h, int n) {
    int q = blockIdx.x * blockDim.x + threadIdx.x;        // one float4 per thread
    long long total = (long long)n * (DIM / 4);
    if (q >= total) return;
    int row = q / (DIM / 4);
    int f4  = q % (DIM / 4);
    int b = batch[row];
    float4 xv = ((const float4*)x)[q];
    float4 qv = ((const float4*)query)[(long long)b * (DIM / 4) + f4];
    float4 r;
    r.x = xv.x * qv.x; r.y = xv.y * qv.y; r.z = xv.z * qv.z; r.w = xv.w * qv.w;
    ((float4*)h)[q] = r;
}

// ---------------------------------------------------------------- A-tile staging
// Stage one 16x128 f32 tile into LDS. Full tiles go through the gfx1250
// async-to-LDS DMA (b128 chunks); partial tiles fall back to guarded copies.
__device__ __forceinline__ void stage_tile(const float* __restrict__ A,
                                           float* __restrict__ buf,
                                           int rowbase, int n_rows,
                                           int tid, int nthreads) {
    if (rowbase + 16 <= n_rows) {
        // 512 float4 chunks, coalesced; LDS addr keeps 16B alignment.
        for (int q = tid; q < 512; q += nthreads) {
            int r  = q >> 5;
            int c4 = (q & 31) << 2;
            const float* g = A + (long long)(rowbase + r) * DIM + c4;
            float*       l = buf + r * LSTRIDE + c4;
#if defined(HAVE_ASYNC_LDS)
            __builtin_amdgcn_global_load_async_to_lds_b128(AS1P(g), AS3P(l), 0, 0);
#else
            *(float4*)l = *(const float4*)g;
#endif
        }
    } else {  // rare ragged tail: guarded scalar copy + zero fill
        for (int j = tid; j < 16 * DIM; j += nthreads) {
            int r = j >> 7, c = j & 127;
            float v = (rowbase + r < n_rows)
                          ? A[(long long)(rowbase + r) * DIM + c] : 0.0f;
            buf[r * LSTRIDE + c] = v;
        }
    }
}

// ---------------------------------------------------------------- WMMA GEMM
// out[N, NO] = A[N, 128] @ W[128, NO].  One wave per 16x16 output tile.
// B fragments (whole W column stripe) live in 64 VGPRs, loaded once per block
// and reused over ROW_ITERS row tiles. A tiles double-buffered in LDS; the
// next tile's async DMA overlaps the current tile's 32 WMMAs.
template <int NO, int ROW_ITERS>
__global__ __launch_bounds__(NO * 2) void gemm_wmma(const float* __restrict__ A,
                                                    const float* __restrict__ W,
                                                    float* __restrict__ out,
                                                    int n_rows) {
    __shared__ float Alds[2][16 * LSTRIDE];
    const int tid  = threadIdx.x;
    const int wave = tid >> 5;
    const int lane = tid & 31;
    const int hi   = lane >> 4;   // half-wave select
    const int lo   = lane & 15;
    const int colbase = wave * 16;

    // Preload B fragments: kk-th covers K = [4kk, 4kk+4).
    // lane l, vgpr j holds B[K = 4kk + 2*hi + j][N = lo].
    v2f bfrag[32];
#pragma unroll
    for (int kk = 0; kk < 32; ++kk) {
        int krow = 4 * kk + 2 * hi;
        bfrag[kk].x = W[krow * NO + colbase + lo];
        bfrag[kk].y = W[(krow + 1) * NO + colbase + lo];
    }

    const int row_tiles = (n_rows + 15) >> 4;
    const int base = blockIdx.x * ROW_ITERS;
    int cur = 0;
    if (base < row_tiles)
        stage_tile(A, Alds[0], base << 4, n_rows, tid, NO * 2);

    for (int it = 0; it < ROW_ITERS; ++it) {
        int rowtile = base + it;
        if (rowtile >= row_tiles) break;               // block-uniform
        int rowbase = rowtile << 4;

        wait_async0();       // our chunks of Alds[cur] have landed
        __syncthreads();     // everyone's chunks have landed, prev reads done

        // Kick off next tile's DMA into the other buffer (safe: that buffer's
        // readers all passed the barrier above).
        if (it + 1 < ROW_ITERS && rowtile + 1 < row_tiles)
            stage_tile(A, Alds[cur ^ 1], (rowtile + 1) << 4, n_rows, tid, NO * 2);

        const float* buf = Alds[cur];
        v8f c = {};
#pragma unroll
        for (int kk = 0; kk < 32; ++kk) {
            int krow = 4 * kk + 2 * hi;
            v2f a;   // lane l, vgpr j holds A[M = lo][K = 4kk + 2*hi + j]
            a.x = buf[lo * LSTRIDE + krow];
            a.y = buf[lo * LSTRIDE + krow + 1];
            c = __builtin_amdgcn_wmma_f32_16x16x4_f32(
                    false, a, false, bfrag[kk], (short)0, c, false, false);
        }

        float* orow = out + (long long)rowbase * NO + colbase;
#pragma unroll
        for (int r = 0; r < 8; ++r) {   // C/D: vgpr r -> row r + 8*hi, col lo
            int row = r + 8 * hi;
            if (rowbase + row < n_rows) orow[row * NO + lo] = c[r];
        }
        cur ^= 1;
    }
}

// ---------------------------------------------------------------- aggregation
// Self-loop term initializes agg (no zero-fill needed): agg_i = hw_i * dis_i^2
__global__ __launch_bounds__(256) void k_selfagg(const float* __restrict__ hw,
                                                 const float* __restrict__ dis,
                                                 float* __restrict__ agg, int n) {
    long long q = (long long)blockIdx.x * blockDim.x + threadIdx.x;
    long long total = (long long)n * (DIM / 4);
    if (q >= total) return;
    int row = (int)(q / (DIM / 4));
    float w = dis[row]; w *= w;
    float4 v = ((const float4*)hw)[q];
    v.x *= w; v.y *= w; v.z *= w; v.w *= w;
    ((float4*)agg)[q] = v;
}

// One wave per edge: lane loads float4 of hw[src] (512B coalesced row) and
// does 4 f32 atomic adds into agg[dst]. src/dst forced into SGPRs.
__global__ __launch_bounds__(256) void k_edgeagg(const int* __restrict__ esrc,
                                                 const int* __restrict__ edst,
                                                 const float* __restrict__ dis,
                                                 const float* __restrict__ hw,
                                                 float* __restrict__ agg, int e) {
    long long gwave = ((long long)blockIdx.x * blockDim.x + threadIdx.x) >> 5;
    int lane = threadIdx.x & 31;
    if (gwave >= e) return;
    int s = __builtin_amdgcn_readfirstlane(esrc[gwave]);   // wave-uniform
    int d = __builtin_amdgcn_readfirstlane(edst[gwave]);
    float w = dis[s] * dis[d];
    float4 v = ((const float4*)(hw + (long long)s * DIM))[lane];
    float* dst = agg + (long long)d * DIM + lane * 4;
    unsafeAtomicAdd(dst + 0, v.x * w);
    unsafeAtomicAdd(dst + 1, v.y * w);
    unsafeAtomicAdd(dst + 2, v.z * w);
    unsafeAtomicAdd(dst + 3, v.w * w);
}

// h_out = [relu](agg + b + res)
__global__ __launch_bounds__(256) void k_finish(const float* __restrict__ agg,
                                                const float* __restrict__ bias,
                                                const float* __restrict__ res,
                                                float* __restrict__ h_out,
                                                int n, int do_relu) {
    long long q = (long long)blockIdx.x * blockDim.x + threadIdx.x;
    long long total = (long long)n * (DIM / 4);
    if (q >= total) return;
    int f4 = (int)(q % (DIM / 4));
    float4 a = ((const float4*)agg)[q];
    float4 b = ((const float4*)bias)[f4];
    float4 r = ((const float4*)res)[q];
    float4 o;
    o.x = a.x + b.x + r.x; o.y = a.y + b.y + r.y;
    o.z = a.z + b.z + r.z; o.w = a.w + b.w + r.w;
    if (do_relu) {
        o.x = fmaxf(o.x, 0.f); o.y = fmaxf(o.y, 0.f);
        o.z = fmaxf(o.z, 0.f); o.w = fmaxf(o.w, 0.f);
    }
    ((float4*)h_out)[q] = o;
}

// out = dropout(pre + bc), p = 0.2 (deterministic hash mask)
__global__ __launch_bounds__(256) void k_out(const float* __restrict__ pre,
                                             const float* __restrict__ bc,
                                             float* __restrict__ out,
                                             long long total, int C) {
    long long i = (long long)blockIdx.x * blockDim.x + threadIdx.x;
    if (i >= total) return;
    int c = (int)(i % C);
    float v = pre[i] + bc[c];
    out[i] = keep_mask((unsigned)i) ? v * 1.25f : 0.0f;
}

// ---------------------------------------------------------------- launcher
extern "C" void kernel_launch(void* const* d_in, const int* in_sizes, int n_in,
                              void* d_out, int out_size, void* d_ws, size_t ws_size,
                              hipStream_t stream) {
    const float* x     = (const float*)d_in[0];
    const float* query = (const float*)d_in[1];
    const int*   batch = (const int*)d_in[2];
    const int*   eidx  = (const int*)d_in[3];
    const float* W0 = (const float*)d_in[4];  const float* b0 = (const float*)d_in[5];
    const float* W1 = (const float*)d_in[6];  const float* b1 = (const float*)d_in[7];
    const float* W2 = (const float*)d_in[8];  const float* b2 = (const float*)d_in[9];
    const float* W3 = (const float*)d_in[10]; const float* b3 = (const float*)d_in[11];
    const float* Wc = (const float*)d_in[12]; const float* bc = (const float*)d_in[13];
    float* out = (float*)d_out;

    const int N = in_sizes[2];
    const int E = in_sizes[3] / 2;
    const int C = in_sizes[13];
    const int* esrc = eidx;       // edge_index[0]
    const int* edst = eidx + E;   // edge_index[1]

    // Workspace carve: dis[N], 3x [N,128] f32 slabs.
    char* ws = (char*)d_ws;
    auto carve = [&](size_t bytes) {
        char* p = ws;
        ws += (bytes + 255) & ~size_t(255);
        return p;
    };
    float* dis  = (float*)carve((size_t)N * 4);
    float* bufA = (float*)carve((size_t)N * DIM * 4);
    float* bufB = (float*)carve((size_t)N * DIM * 4);
    float* bufC = (float*)carve((size_t)N * DIM * 4);
    (void)ws_size;

    const int T = 256;
    const long long nd4 = (long long)N * (DIM / 4);
    dim3 gN((N + T - 1) / T), gE((E + T - 1) / T);
    dim3 gND((unsigned)((nd4 + T - 1) / T));
    dim3 gEW((unsigned)(((long long)E * 32 + T - 1) / T));  // one wave per edge

    // norm
    k_fill1<<<gN, T, 0, stream>>>(dis, N);
    k_deg<<<gE, T, 0, stream>>>(edst, dis, E);
    k_rsqrt<<<gN, T, 0, stream>>>(dis, N);

    // h0 = query[batch] * x  -> bufA
    k_inith<<<gND, T, 0, stream>>>(x, query, batch, bufA, N);

    const int rowTiles = (N + 15) / 16;
    const int RI = 10;
    dim3 gG((rowTiles + RI - 1) / RI);

    // Ping-pong: layer reads hIn, gemm->hw, agg->bufC, finish->hOut (= hw slab).
    const float* Ws[4]   = {W0, W1, W2, W3};
    const float* bs[4]   = {b0, b1, b2, b3};
    float* hIn = bufA;
    float* hw  = bufB;
    for (int l = 0; l < 4; ++l) {
        gemm_wmma<DIM, RI><<<gG, DIM * 2, 0, stream>>>(hIn, Ws[l], hw, N);
        k_selfagg<<<gND, T, 0, stream>>>(hw, dis, bufC, N);
        k_edgeagg<<<gEW, T, 0, stream>>>(esrc, edst, dis, hw, bufC, E);
        k_finish<<<gND, T, 0, stream>>>(bufC, bs[l], hIn, hw, N, l < 3 ? 1 : 0);
        // swap: new h lives in hw slab; old hIn slab becomes next hw scratch
        float* t = hIn; hIn = hw; hw = t;
    }

    // Final: pre = h @ Wc -> hw slab (N x 64 fits), then bias + dropout -> out
    gemm_wmma<64, RI><<<gG, 64 * 2, 0, stream>>>(hIn, Wc, hw, N);
    long long totalOut = (long long)N * C;
    dim3 gO((unsigned)((totalOut + T - 1) / T));
    k_out<<<gO, T, 0, stream>>>(hw, bc, out, totalOut, C);
}